// TCN_cell_40982577938734
// MI455X (gfx1250) — compile-verified
//
#include <hip/hip_runtime.h>
#include <hip/hip_bf16.h>

// Spatio-temporal GNN (EmbGCN x3 -> dilated TCN x3 -> temporal attention x2)
// for MI455X / gfx1250. All GEMM stages run on V_WMMA_F32_16X16X4_F32
// (exact fp32, matches reference numerics) with the ISA-documented f32
// operand layouts (CDNA5 ISA 7.12.2). Data staging uses the
// CDNA5 async paths: TDM tensor_load_to_lds (TENSORcnt) for the 64KB
// H-panel in the dominant spatial GEMM, and global_load_async_to_lds_b128
// (ASYNCcnt) for the attention h-slice.

typedef float v2f __attribute__((ext_vector_type(2)));
typedef float v8f __attribute__((ext_vector_type(8)));
typedef unsigned int u32x4 __attribute__((ext_vector_type(4)));
typedef int i32x4 __attribute__((ext_vector_type(4)));
typedef int i32x8 __attribute__((ext_vector_type(8)));

__device__ __forceinline__ v8f wmma_f32(v2f a, v2f b, v8f c) {
  // 8-arg pattern: (neg_a, A, neg_b, B, c_mod, C, reuse_a, reuse_b)
  return __builtin_amdgcn_wmma_f32_16x16x4_f32(false, a, false, b, (short)0, c,
                                               false, false);
}

// ---------------------------------------------------------------------------
// Sizes
// ---------------------------------------------------------------------------
#define NB 4
#define NT 128
#define NN 512
#define ND 32
#define NBT 512   // NB*NT
#define NBN 2048  // NB*NN

// ---------------------------------------------------------------------------
// S2 = 2*adj@adj - I   (Chebyshev support k=2). WMMA 512x512x512 GEMM.
// ---------------------------------------------------------------------------
__global__ void k_cheb(const float* __restrict__ adj, float* __restrict__ s2) {
  const int lane = threadIdx.x & 31, wave = threadIdx.x >> 5;
  const int lo = lane & 15, hi = lane >> 4;
  const int rowBase = blockIdx.x * 16;
  const int colBase = (blockIdx.y * 4 + wave) * 32;
  v8f acc0 = {}, acc1 = {};
  for (int kk = 0; kk < NN; kk += 4) {
    const float* ap = adj + (rowBase + lo) * NN + kk + 2 * hi;
    v2f a; a.x = ap[0]; a.y = ap[1];
    const float* bp = adj + (kk + 2 * hi) * NN + colBase + lo;
    v2f b0, b1;
    b0.x = bp[0];  b0.y = bp[NN];
    b1.x = bp[16]; b1.y = bp[NN + 16];
    acc0 = wmma_f32(a, b0, acc0);
    acc1 = wmma_f32(a, b1, acc1);
  }
  for (int r = 0; r < 8; ++r) {
    int row = rowBase + hi * 8 + r;
    int c0 = colBase + lo, c1 = colBase + 16 + lo;
    s2[row * NN + c0] = 2.0f * acc0[r] - (row == c0 ? 1.0f : 0.0f);
    s2[row * NN + c1] = 2.0f * acc1[r] - (row == c1 ? 1.0f : 0.0f);
  }
}

// ---------------------------------------------------------------------------
// Per-node GCN weights: wn[n][k][c][o] = sum_e E[n][e]*wp[e][k][c][o]
//                       bn[n][o]       = sum_e E[n][e]*bp[e][o]
// ---------------------------------------------------------------------------
__global__ void k_nodew(const float* __restrict__ E, const float* __restrict__ wp,
                        const float* __restrict__ bp, float* __restrict__ wn,
                        float* __restrict__ bn) {
  int idx = blockIdx.x * blockDim.x + threadIdx.x;
  if (idx < NN * 3 * ND * ND) {
    int o = idx & 31, c = (idx >> 5) & 31, k = (idx >> 10) % 3, n = idx / (3 * ND * ND);
    float s = 0.f;
    for (int e = 0; e < 10; ++e) s += E[n * 10 + e] * wp[((e * 3 + k) * ND + c) * ND + o];
    wn[idx] = s;
  }
  if (idx < NN * ND) {
    int o = idx & 31, n = idx >> 5;
    float s = 0.f;
    for (int e = 0; e < 10; ++e) s += E[n * 10 + e] * bp[e * ND + o];
    bn[idx] = s;
  }
}

// ---------------------------------------------------------------------------
// Spatial mixing: G[bt] = S @ H[bt], S (512x512), H (512x32).
// Block per bt, 512 threads (16 waves). The 64KB H panel is DMA'd into LDS
// by the Tensor Data Mover (one descriptor, issued by wave 0, TENSORcnt).
// ---------------------------------------------------------------------------
__global__ void k_spatial(const float* __restrict__ S, const float* __restrict__ h,
                          float* __restrict__ g) {
  extern __shared__ float Hs[];  // 512*32 floats
  const int bt = blockIdx.x;
  const float* hb = h + (size_t)bt * NN * ND;

  if (threadIdx.x < 32) {  // wave 0 issues the TDM load of H (512x32 f32)
    unsigned long long ga = (unsigned long long)(uintptr_t)hb;
    unsigned int la = (unsigned int)(uintptr_t)Hs;  // wave-relative LDS offset
    u32x4 g0;
    g0.x = 1u;                    // count=1, user descriptor
    g0.y = la;                    // lds_addr
    g0.z = (unsigned int)ga;      // global_addr[31:0]
    g0.w = ((unsigned int)(ga >> 32) & 0x01FFFFFFu) | (2u << 30);  // addr[56:32], type=2
    i32x8 g1;
    g1[0] = (int)(2u << 16);      // data_size = 4B (code 2); wg_mask=0
    g1[1] = (int)(32u << 16);     // tensor_dim0 = 32 (low 16 bits)
    g1[2] = (int)(512u << 16);    // tensor_dim0 hi=0 | tensor_dim1 = 512
    g1[3] = (int)(32u << 16);     // tensor_dim1 hi=0 | tile_dim0 = 32
    g1[4] = 512;                  // tile_dim1 = 512, tile_dim2 = 0
    g1[5] = 32;                   // tensor_dim0_stride = 32
    g1[6] = 0;                    // stride0 hi | stride1 lo
    g1[7] = 0;                    // stride1 hi
    i32x4 gz = {0, 0, 0, 0};
#if __clang_major__ >= 23
    i32x8 gz8 = {0, 0, 0, 0, 0, 0, 0, 0};
    __builtin_amdgcn_tensor_load_to_lds(g0, g1, gz, gz, gz8, 0);
#else
    __builtin_amdgcn_tensor_load_to_lds(g0, g1, gz, gz, 0);
#endif
    __builtin_amdgcn_s_wait_tensorcnt(0);
  }
  __syncthreads();

  const int lane = threadIdx.x & 31, wave = threadIdx.x >> 5;
  const int lo = lane & 15, hi = lane >> 4;
  float* gb = g + (size_t)bt * NN * ND;
  for (int job = wave; job < 32; job += 16) {
    const int rowBase = job * 16;
    v8f acc0 = {}, acc1 = {};
    for (int kk = 0; kk < NN; kk += 4) {
      const float* ap = S + (rowBase + lo) * NN + kk + 2 * hi;
      v2f a; a.x = ap[0]; a.y = ap[1];
      const float* bp = Hs + (kk + 2 * hi) * ND + lo;
      v2f b0, b1;
      b0.x = bp[0];  b0.y = bp[ND];
      b1.x = bp[16]; b1.y = bp[ND + 16];
      acc0 = wmma_f32(a, b0, acc0);
      acc1 = wmma_f32(a, b1, acc1);
    }
    for (int r = 0; r < 8; ++r) {
      int row = rowBase + hi * 8 + r;
      gb[row * ND + lo] = acc0[r];
      gb[row * ND + 16 + lo] = acc1[r];
    }
  }
}

// ---------------------------------------------------------------------------
// Per-node GEMM: out[bt,n,:] (+)= G[bt,n,:] @ wn[n][k] (+ bn[n] when k==0).
// ---------------------------------------------------------------------------
__global__ void k_nodegemm(const float* __restrict__ G, const float* __restrict__ wn,
                           const float* __restrict__ bn, float* __restrict__ out,
                           int k, int initBias) {
  __shared__ float Ws[ND * ND];
  const int n = blockIdx.x;
  const float* wsrc = wn + ((size_t)n * 3 + k) * ND * ND;
  for (int i = threadIdx.x; i < ND * ND; i += blockDim.x) Ws[i] = wsrc[i];
  __syncthreads();
  const int lane = threadIdx.x & 31, wave = threadIdx.x >> 5;
  const int lo = lane & 15, hi = lane >> 4;
  const int btBase = (blockIdx.y * 4 + wave) * 16;
  v8f acc0, acc1;
  if (initBias) {
    float b0 = bn[n * ND + lo], b1 = bn[n * ND + 16 + lo];
    for (int r = 0; r < 8; ++r) { acc0[r] = b0; acc1[r] = b1; }
  } else {
    for (int r = 0; r < 8; ++r) {
      size_t row = (size_t)(btBase + hi * 8 + r) * NN + n;
      acc0[r] = out[row * ND + lo];
      acc1[r] = out[row * ND + 16 + lo];
    }
  }
  for (int kk = 0; kk < ND; kk += 4) {
    const float* ap = G + ((size_t)(btBase + lo) * NN + n) * ND + kk + 2 * hi;
    v2f a; a.x = ap[0]; a.y = ap[1];
    const float* bp = Ws + (kk + 2 * hi) * ND + lo;
    v2f b0, b1;
    b0.x = bp[0];  b0.y = bp[ND];
    b1.x = bp[16]; b1.y = bp[ND + 16];
    acc0 = wmma_f32(a, b0, acc0);
    acc1 = wmma_f32(a, b1, acc1);
  }
  for (int r = 0; r < 8; ++r) {
    size_t row = (size_t)(btBase + hi * 8 + r) * NN + n;
    out[row * ND + lo] = acc0[r];
    out[row * ND + 16 + lo] = acc1[r];
  }
}

// ---------------------------------------------------------------------------
// TCN weight transpose: wt[conv][o][q = j*32 + c] from (blk, O, C, K)
// ---------------------------------------------------------------------------
__global__ void k_tcnprep(const float* __restrict__ w1, const float* __restrict__ w2,
                          float* __restrict__ wt) {
  int idx = blockIdx.x * blockDim.x + threadIdx.x;
  if (idx >= 6 * ND * 96) return;
  int q = idx % 96, o = (idx / 96) % ND, conv = idx / (96 * ND);
  int blk = conv >> 1, which = conv & 1;
  int j = q >> 5, c = q & 31;
  const float* w = which ? w2 : w1;
  wt[idx] = w[(((size_t)blk * ND + o) * ND + c) * 3 + j];
}

__device__ __forceinline__ float tcn_b_elem(const float* __restrict__ src, int q,
                                            int t, int dil) {
  int j = q >> 5, c = q & 31;
  int tt = t - (2 - j) * dil;
  return (tt >= 0) ? src[c * NT + tt] : 0.0f;
}

// One causal conv as (32 x 96) @ (96 x 128) WMMA GEMM on LDS-resident data.
__device__ void tcn_conv(const float* __restrict__ src, float* __restrict__ dst,
                         const float* __restrict__ wt, const float* __restrict__ bias,
                         int dil, const float* __restrict__ residual, int tid) {
  const int lane = tid & 31, wave = tid >> 5;
  const int lo = lane & 15, hi = lane >> 4;
  for (int job = wave; job < 8; job += 4) {
    const int rowBase = (job & 1) * 16;
    const int colBase = (job >> 1) * 32;
    v8f acc0, acc1;
    for (int r = 0; r < 8; ++r) {
      float bv = bias[rowBase + hi * 8 + r];
      acc0[r] = bv; acc1[r] = bv;
    }
    for (int kk = 0; kk < 96; kk += 4) {
      const int q0 = kk + 2 * hi;
      const float* ap = wt + (rowBase + lo) * 96 + q0;
      v2f a; a.x = ap[0]; a.y = ap[1];
      v2f b0, b1;
      b0.x = tcn_b_elem(src, q0, colBase + lo, dil);
      b0.y = tcn_b_elem(src, q0 + 1, colBase + lo, dil);
      b1.x = tcn_b_elem(src, q0, colBase + 16 + lo, dil);
      b1.y = tcn_b_elem(src, q0 + 1, colBase + 16 + lo, dil);
      acc0 = wmma_f32(a, b0, acc0);
      acc1 = wmma_f32(a, b1, acc1);
    }
    for (int r = 0; r < 8; ++r) {
      int o = rowBase + hi * 8 + r;
      int t0 = colBase + lo, t1 = colBase + 16 + lo;
      float v0 = fmaxf(acc0[r], 0.f), v1 = fmaxf(acc1[r], 0.f);
      if (residual) {
        v0 = fmaxf(v0 + residual[o * NT + t0], 0.f);
        v1 = fmaxf(v1 + residual[o * NT + t1], 0.f);
      }
      dst[o * NT + t0] = v0;
      dst[o * NT + t1] = v1;
    }
  }
}

// Block per sequence (b,n). Loads (D,T) slice into LDS, runs 3 temporal
// blocks (dil 1,2,4), writes (seq,t,c)-layout output for attention.
__global__ void k_tcn(const float* __restrict__ h, float* __restrict__ h2,
                      const float* __restrict__ wt, const float* __restrict__ b1,
                      const float* __restrict__ b2) {
  __shared__ float ys[ND * NT];
  __shared__ float ts1[ND * NT];
  const int seq = blockIdx.x;
  const int b = seq >> 9, n = seq & 511;
  for (int t = threadIdx.x; t < NT; t += blockDim.x) {
    const float* src = h + ((size_t)(b * NT + t) * NN + n) * ND;
    for (int c = 0; c < ND; ++c) ys[c * NT + t] = src[c];
  }
  __syncthreads();
  for (int blk = 0; blk < 3; ++blk) {
    const int dil = 1 << blk;
    tcn_conv(ys, ts1, wt + (blk * 2) * ND * 96, b1 + blk * ND, dil, nullptr,
             threadIdx.x);
    __syncthreads();
    tcn_conv(ts1, ys, wt + (blk * 2 + 1) * ND * 96, b2 + blk * ND, dil, ys,
             threadIdx.x);
    __syncthreads();
  }
  for (int t = threadIdx.x; t < NT; t += blockDim.x) {
    float* dst = h2 + ((size_t)seq * NT + t) * ND;
    for (int c = 0; c < ND; ++c) dst[c] = ys[c * NT + t];
  }
}

// ---------------------------------------------------------------------------
// Attention core: block per (seq, head). QKV proj, QK^T, softmax, PV — all
// WMMA on LDS-resident tiles (104KB dynamic LDS, under WGP's 320KB).
// h-slice staged via async global->LDS (ASYNCcnt), no VGPR round-trip.
// ---------------------------------------------------------------------------
__global__ void k_attn_core(const float* __restrict__ h2, const float* __restrict__ wq,
                            const float* __restrict__ wk, const float* __restrict__ wv,
                            float* __restrict__ attnO) {
  extern __shared__ float lds[];
  float* hs = lds;                 // 128*32
  float* qs = hs + NT * ND;        // 128*16
  float* ks = qs + NT * 16;
  float* vs = ks + NT * 16;
  float* ss = vs + NT * 16;        // 128*128
  const int seq = blockIdx.x, head = blockIdx.y;
  const float* src = h2 + (size_t)seq * NT * ND;
  // async copy 16KB h-slice into LDS: 1024 x B128, per-lane LDS dest addrs
  for (int i = threadIdx.x; i < NT * ND / 4; i += blockDim.x) {
    unsigned int la = (unsigned int)(uintptr_t)hs + (unsigned int)i * 16u;
    unsigned long long ga = (unsigned long long)(uintptr_t)src + (unsigned long long)i * 16u;
    asm volatile("global_load_async_to_lds_b128 %0, %1, off"
                 :: "v"(la), "v"(ga) : "memory");
  }
  asm volatile("s_wait_asynccnt 0x0" ::: "memory");
  __syncthreads();
  const int lane = threadIdx.x & 31, wave = threadIdx.x >> 5;
  const int lo = lane & 15, hi = lane >> 4;
  // Q/K/V head slices: (128x32)@(32x16), 24 tile jobs
  for (int job = wave; job < 24; job += 4) {
    const int mat = job >> 3, rt = job & 7;
    const float* w = (mat == 0) ? wq : (mat == 1) ? wk : wv;
    float* dst = (mat == 0) ? qs : (mat == 1) ? ks : vs;
    v8f acc = {};
    for (int kk = 0; kk < ND; kk += 4) {
      const float* ap = hs + (rt * 16 + lo) * ND + kk + 2 * hi;
      v2f a; a.x = ap[0]; a.y = ap[1];
      const float* bp = w + (kk + 2 * hi) * ND + head * 16 + lo;
      v2f b; b.x = bp[0]; b.y = bp[ND];
      acc = wmma_f32(a, b, acc);
    }
    for (int r = 0; r < 8; ++r) dst[(rt * 16 + hi * 8 + r) * 16 + lo] = acc[r];
  }
  __syncthreads();
  // scores = (Q @ K^T) * 1/sqrt(16)
  const float scale = 0.25f;
  for (int job = wave; job < 64; job += 4) {
    const int rt = job >> 3, ct = job & 7;
    v8f acc = {};
    for (int kk = 0; kk < 16; kk += 4) {
      const float* ap = qs + (rt * 16 + lo) * 16 + kk + 2 * hi;
      v2f a; a.x = ap[0]; a.y = ap[1];
      const float* bp = ks + (ct * 16 + lo) * 16 + kk + 2 * hi;
      v2f b; b.x = bp[0]; b.y = bp[1];
      acc = wmma_f32(a, b, acc);
    }
    for (int r = 0; r < 8; ++r)
      ss[(rt * 16 + hi * 8 + r) * NT + ct * 16 + lo] = acc[r] * scale;
  }
  __syncthreads();
  // softmax per row (one thread per query position)
  {
    float* p = ss + threadIdx.x * NT;
    float m = p[0];
    for (int i = 1; i < NT; ++i) m = fmaxf(m, p[i]);
    float s = 0.f;
    for (int i = 0; i < NT; ++i) { float e = __expf(p[i] - m); p[i] = e; s += e; }
    float inv = 1.f / s;
    for (int i = 0; i < NT; ++i) p[i] *= inv;
  }
  __syncthreads();
  // O_head = P @ V : (128x128)@(128x16)
  for (int rt = wave; rt < 8; rt += 4) {
    v8f acc = {};
    for (int kk = 0; kk < NT; kk += 4) {
      const float* ap = ss + (rt * 16 + lo) * NT + kk + 2 * hi;
      v2f a; a.x = ap[0]; a.y = ap[1];
      const float* bp = vs + (kk + 2 * hi) * 16 + lo;
      v2f b; b.x = bp[0]; b.y = bp[16];
      acc = wmma_f32(a, b, acc);
    }
    for (int r = 0; r < 8; ++r)
      attnO[((size_t)seq * NT + rt * 16 + hi * 8 + r) * ND + head * 16 + lo] = acc[r];
  }
}

// ---------------------------------------------------------------------------
// Output projection + residual + LayerNorm, in-place on h2. Block per seq.
// ---------------------------------------------------------------------------
__global__ void k_attn_proj_ln(float* __restrict__ h2, const float* __restrict__ attnO,
                               const float* __restrict__ wo, const float* __restrict__ g,
                               const float* __restrict__ bb) {
  __shared__ float as[NT * ND];
  __shared__ float rs[NT * ND];
  __shared__ float wos[ND * ND];
  const int seq = blockIdx.x;
  const float* src = attnO + (size_t)seq * NT * ND;
  for (int i = threadIdx.x; i < NT * ND / 4; i += blockDim.x)
    ((float4*)as)[i] = ((const float4*)src)[i];
  for (int i = threadIdx.x; i < ND * ND; i += blockDim.x) wos[i] = wo[i];
  __syncthreads();
  const int lane = threadIdx.x & 31, wave = threadIdx.x >> 5;
  const int lo = lane & 15, hi = lane >> 4;
  float* hseq = h2 + (size_t)seq * NT * ND;
  for (int rt = wave; rt < 8; rt += 4) {
    const int rowBase = rt * 16;
    v8f acc0 = {}, acc1 = {};
    for (int kk = 0; kk < ND; kk += 4) {
      const float* ap = as + (rowBase + lo) * ND + kk + 2 * hi;
      v2f a; a.x = ap[0]; a.y = ap[1];
      const float* bp = wos + (kk + 2 * hi) * ND + lo;
      v2f b0, b1;
      b0.x = bp[0];  b0.y = bp[ND];
      b1.x = bp[16]; b1.y = bp[ND + 16];
      acc0 = wmma_f32(a, b0, acc0);
      acc1 = wmma_f32(a, b1, acc1);
    }
    for (int r = 0; r < 8; ++r) {
      int t = rowBase + hi * 8 + r;
      rs[t * ND + lo] = acc0[r] + hseq[t * ND + lo];
      rs[t * ND + 16 + lo] = acc1[r] + hseq[t * ND + 16 + lo];
    }
  }
  __syncthreads();
  {
    const int t = threadIdx.x;
    float* p = rs + t * ND;
    float mu = 0.f;
    for (int i = 0; i < ND; ++i) mu += p[i];
    mu *= (1.0f / ND);
    float var = 0.f;
    for (int i = 0; i < ND; ++i) { float d = p[i] - mu; var += d * d; }
    var *= (1.0f / ND);
    float inv = rsqrtf(var + 1e-5f);
    float* dst = hseq + t * ND;
    for (int i = 0; i < ND; ++i) dst[i] = (p[i] - mu) * inv * g[i] + bb[i];
  }
}

// (seq,t,d) -> (b,t,n,d)
__global__ void k_final(const float* __restrict__ h2, float* __restrict__ out) {
  size_t idx = (size_t)blockIdx.x * blockDim.x + threadIdx.x;
  if (idx >= (size_t)NB * NT * NN * ND) return;
  int d = idx & 31;
  int n = (idx >> 5) & 511;
  int t = (idx >> 14) & 127;
  int b = idx >> 21;
  out[idx] = h2[(((size_t)(b * NN + n)) * NT + t) * ND + d];
}

// ---------------------------------------------------------------------------
extern "C" void kernel_launch(void* const* d_in, const int* in_sizes, int n_in,
                              void* d_out, int out_size, void* d_ws, size_t ws_size,
                              hipStream_t stream) {
  const float* x     = (const float*)d_in[0];
  const float* E     = (const float*)d_in[2];
  const float* adj   = (const float*)d_in[3];
  const float* gcn_w = (const float*)d_in[4];   // (3,10,3,32,32)
  const float* gcn_b = (const float*)d_in[5];   // (3,10,32)
  const float* tw1   = (const float*)d_in[6];   // (3,32,32,3)
  const float* tb1   = (const float*)d_in[7];
  const float* tw2   = (const float*)d_in[8];
  const float* tb2   = (const float*)d_in[9];
  const float* wq    = (const float*)d_in[10];  // (2,32,32)
  const float* wk    = (const float*)d_in[11];
  const float* wv    = (const float*)d_in[12];
  const float* wo    = (const float*)d_in[13];
  const float* tg    = (const float*)d_in[14];  // (2,32)
  const float* tbb   = (const float*)d_in[15];

  // Workspace carve-up (floats). Total ~108.1 MB.
  float* ws = (float*)d_ws;
  float* s2   = ws;                                  // 512*512
  float* wn   = s2 + (size_t)NN * NN;                // 512*3*32*32
  float* bn   = wn + (size_t)NN * 3 * ND * ND;       // 512*32
  float* wt   = bn + (size_t)NN * ND;                // 6*32*96
  float* hA   = wt + 6 * ND * 96;                    // 512*512*32
  float* hB   = hA + (size_t)NBT * NN * ND;          // 512*512*32
  float* gbuf = hB + (size_t)NBT * NN * ND;          // 512*512*32 (also attnO)

  k_cheb<<<dim3(32, 4), 128, 0, stream>>>(adj, s2);
  k_tcnprep<<<(6 * ND * 96 + 255) / 256, 256, 0, stream>>>(tw1, tw2, wt);

  const float* hin = x;
  float* hbuf[2] = {hA, hB};
  for (int layer = 0; layer < 3; ++layer) {
    float* hout = hbuf[layer & 1];
    k_nodew<<<(NN * 3 * ND * ND + 255) / 256, 256, 0, stream>>>(
        E, gcn_w + (size_t)layer * 10 * 3 * ND * ND, gcn_b + (size_t)layer * 10 * ND,
        wn, bn);
    // k=0: G = h (identity support), seed with bias
    k_nodegemm<<<dim3(NN, 8), 128, 0, stream>>>(hin, wn, bn, hout, 0, 1);
    // k=1: G = adj @ h
    k_spatial<<<NBT, 512, NN * ND * 4, stream>>>(adj, hin, gbuf);
    k_nodegemm<<<dim3(NN, 8), 128, 0, stream>>>(gbuf, wn, bn, hout, 1, 0);
    // k=2: G = (2 adj^2 - I) @ h
    k_spatial<<<NBT, 512, NN * ND * 4, stream>>>(s2, hin, gbuf);
    k_nodegemm<<<dim3(NN, 8), 128, 0, stream>>>(gbuf, wn, bn, hout, 2, 0);
    hin = hout;
  }

  // TCN: (bt,n,c) -> (seq,t,c) with 3 residual temporal blocks
  k_tcn<<<NBN, 128, 0, stream>>>(hin, hB, wt, tb1, tb2);

  // Temporal attention x2, in-place on hB
  const size_t attnLds = (size_t)(NT * ND + 3 * NT * 16 + NT * NT) * 4;
  for (int l = 0; l < 2; ++l) {
    k_attn_core<<<dim3(NBN, 2), 128, attnLds, stream>>>(
        hB, wq + l * ND * ND, wk + l * ND * ND, wv + l * ND * ND, gbuf);
    k_attn_proj_ln<<<NBN, 128, 0, stream>>>(hB, gbuf, wo + l * ND * ND,
                                            tg + l * ND, tbb + l * ND);
  }

  k_final<<<(NB * NT * NN * ND + 255) / 256, 256, 0, stream>>>(hB, (float*)d_out);
}